// BaseDepthVolumeModel_6468220748328
// MI455X (gfx1250) — compile-verified
//
#include <hip/hip_runtime.h>

typedef float v2f __attribute__((ext_vector_type(2)));
typedef float v8f __attribute__((ext_vector_type(8)));

#define NS 2
#define NV 4
#define H_ 256
#define W_ 320
#define ND 64
#define DEPTH0 0.5f
#define DEPTH_STEP (9.5f / 63.0f)

__device__ __forceinline__ void inv3x3(const float* m, float* o) {
    float c00 = m[4]*m[8] - m[5]*m[7];
    float c10 = m[5]*m[6] - m[3]*m[8];
    float c20 = m[3]*m[7] - m[4]*m[6];
    float det = m[0]*c00 + m[1]*c10 + m[2]*c20;
    float id  = 1.0f / det;
    o[0] = c00 * id;
    o[1] = (m[2]*m[7] - m[1]*m[8]) * id;
    o[2] = (m[1]*m[5] - m[2]*m[4]) * id;
    o[3] = c10 * id;
    o[4] = (m[0]*m[8] - m[2]*m[6]) * id;
    o[5] = (m[2]*m[3] - m[0]*m[5]) * id;
    o[6] = c20 * id;
    o[7] = (m[1]*m[6] - m[0]*m[7]) * id;
    o[8] = (m[0]*m[4] - m[1]*m[3]) * id;
}

// One wave. Computes per-(n,v): M3 = srcK * R * inv(Kd) and b3 = srcK * t,
// where [R|t] = srcE * inv(dstE). The batched 4x4 composition T = srcE*invE
// is done with V_WMMA_F32_16X16X4_F32: A = 16x4 stack of the 4 view matrices
// (rows 4v+i), B = invE in columns 0..3 -> D columns 0..3 hold T per view.
__global__ __launch_bounds__(32) void setup_kernel(
    const float* __restrict__ dstK, const float* __restrict__ dstE,
    const float* __restrict__ srcK, const float* __restrict__ srcE,
    float* __restrict__ ws)
{
    __shared__ float Tl[NS * NV * 16];
    const int lane = threadIdx.x;
    const int mrow = lane & 15;   // A-matrix row (= 4*view + i), also B column
    const int half = lane >> 4;
    const int k0   = half * 2;    // K indices held by this lane: k0, k0+1

    for (int n = 0; n < NS; ++n) {
        // invE of dstE[n] (last row is [0,0,0,1] by construction)
        float E[16];
        #pragma unroll
        for (int i = 0; i < 16; ++i) E[i] = dstE[n*16 + i];
        float R[9] = { E[0],E[1],E[2], E[4],E[5],E[6], E[8],E[9],E[10] };
        float Ri[9];
        inv3x3(R, Ri);
        float iE[16];
        iE[0]=Ri[0]; iE[1]=Ri[1]; iE[2]=Ri[2];
        iE[4]=Ri[3]; iE[5]=Ri[4]; iE[6]=Ri[5];
        iE[8]=Ri[6]; iE[9]=Ri[7]; iE[10]=Ri[8];
        iE[3]  = -(Ri[0]*E[3] + Ri[1]*E[7] + Ri[2]*E[11]);
        iE[7]  = -(Ri[3]*E[3] + Ri[4]*E[7] + Ri[5]*E[11]);
        iE[11] = -(Ri[6]*E[3] + Ri[7]*E[7] + Ri[8]*E[11]);
        iE[12] = 0.0f; iE[13] = 0.0f; iE[14] = 0.0f; iE[15] = 1.0f;

        // A regs: A[mrow][k0], A[mrow][k0+1]  (16x4 f32 layout)
        const int va = mrow >> 2, ia = mrow & 3;
        const float* Erow = srcE + (((n*NV + va)*4 + ia) * 4);
        v2f a; a.x = Erow[k0]; a.y = Erow[k0 + 1];
        // B regs: B[k0+j][col], col = mrow (cols >= 4 unused -> 0)
        v2f b;
        b.x = (mrow < 4) ? iE[(k0+0)*4 + mrow] : 0.0f;
        b.y = (mrow < 4) ? iE[(k0+1)*4 + mrow] : 0.0f;
        v8f c = {};
        v8f d = __builtin_amdgcn_wmma_f32_16x16x4_f32(
            false, a, false, b, (short)0, c, false, false);

        // D element (M,N): VGPR = M&7, lane = N + 16*(M>>3)
        if (mrow < 4) {
            #pragma unroll
            for (int j = 0; j < 8; ++j) {
                int M = j + 8*half;
                Tl[(n*NV + (M >> 2))*16 + (M & 3)*4 + mrow] = d[j];
            }
        }
    }
    __syncthreads();

    if (lane < NS * NV) {
        const int n = lane >> 2, v = lane & 3;
        const float* Tv = Tl + (n*NV + v) * 16;
        float K[9];
        #pragma unroll
        for (int i = 0; i < 9; ++i) K[i] = srcK[(n*NV + v)*9 + i];
        float A3[9], b3[3];
        #pragma unroll
        for (int i = 0; i < 3; ++i) {
            #pragma unroll
            for (int j = 0; j < 3; ++j)
                A3[i*3+j] = K[i*3+0]*Tv[0*4+j] + K[i*3+1]*Tv[1*4+j] + K[i*3+2]*Tv[2*4+j];
            b3[i] = K[i*3+0]*Tv[3] + K[i*3+1]*Tv[7] + K[i*3+2]*Tv[11];
        }
        float Kd[9], iKd[9];
        #pragma unroll
        for (int i = 0; i < 9; ++i) Kd[i] = dstK[n*9 + i];
        inv3x3(Kd, iKd);
        float* o = ws + lane * 12;
        #pragma unroll
        for (int i = 0; i < 3; ++i)
            #pragma unroll
            for (int j = 0; j < 3; ++j)
                o[i*3+j] = A3[i*3+0]*iKd[0*3+j] + A3[i*3+1]*iKd[1*3+j] + A3[i*3+2]*iKd[2*3+j];
        o[9] = b3[0]; o[10] = b3[1]; o[11] = b3[2];
    }
}

// Streaming kernel: one thread per pixel of one (n,v); loops all 64 depths.
// Pure store-bandwidth bound (503 MB out) -> non-temporal stores.
__global__ __launch_bounds__(256) void sweep_kernel(
    const float* __restrict__ ws, float* __restrict__ out)
{
    const int g = blockIdx.y;                              // n*NV + v
    const int p = blockIdx.x * 256 + threadIdx.x;          // pixel index in HxW
    const float* P = ws + g * 12;                          // uniform -> scalar loads
    const float m00 = P[0], m01 = P[1], m02 = P[2];
    const float m10 = P[3], m11 = P[4], m12 = P[5];
    const float m20 = P[6], m21 = P[7], m22 = P[8];
    const float b0  = P[9], b1  = P[10], b2 = P[11];

    const float x = (float)(p % W_);
    const float y = (float)(p / W_);
    const float bx = fmaf(m00, x, fmaf(m01, y, m02));
    const float by = fmaf(m10, x, fmaf(m11, y, m12));
    const float bz = fmaf(m20, x, fmaf(m21, y, m22));

    const size_t HW = (size_t)H_ * (size_t)W_;
    v2f*   xy_out   = (v2f*)out;
    float* mask_out = out + (size_t)NS * NV * ND * HW * 2;

    #pragma unroll 4
    for (int d = 0; d < ND; ++d) {
        const float depth = fmaf((float)d, DEPTH_STEP, DEPTH0);
        const float px = fmaf(depth, bx, b0);
        const float py = fmaf(depth, by, b1);
        const float pz = fmaf(depth, bz, b2);
        const float zs = (fabsf(pz) < 1e-8f) ? 1e-8f : pz;
        const float xx = px / zs;
        const float yy = py / zs;
        const float mk = (xx >= 0.0f && xx <= (float)(W_ - 1) &&
                          yy >= 0.0f && yy <= (float)(H_ - 1) &&
                          pz > 0.0f) ? 1.0f : 0.0f;
        const size_t idx = ((size_t)(g * ND + d)) * HW + (size_t)p;
        v2f xy; xy.x = xx; xy.y = yy;
        __builtin_nontemporal_store(xy, xy_out + idx);
        __builtin_nontemporal_store(mk, mask_out + idx);
    }
}

extern "C" void kernel_launch(void* const* d_in, const int* in_sizes, int n_in,
                              void* d_out, int out_size, void* d_ws, size_t ws_size,
                              hipStream_t stream) {
    const float* dstK = (const float*)d_in[0];   // (N,1,3,3)
    const float* dstE = (const float*)d_in[1];   // (N,1,4,4)
    const float* srcK = (const float*)d_in[2];   // (N,V,3,3)
    const float* srcE = (const float*)d_in[3];   // (N,V,4,4)
    float* ws = (float*)d_ws;

    setup_kernel<<<1, 32, 0, stream>>>(dstK, dstE, srcK, srcE, ws);

    dim3 grid((H_ * W_) / 256, NS * NV);
    sweep_kernel<<<grid, 256, 0, stream>>>(ws, (float*)d_out);
}